// BundleAdjustment_33268816674950
// MI455X (gfx1250) — compile-verified
//
#include <hip/hip_runtime.h>
#include <math.h>

namespace {

constexpr int   P_NUM  = 2048;
constexpr int   POSE_F = 7;                 // t(3) + q(4)
constexpr int   E_NUM  = 2048 * 1024;       // 2,097,152
constexpr float FOV_H  = 2.0943951023931953f;
constexpr float R_SCALE = 512.0f / (30.0f - 0.1f);
constexpr float T_SCALE = 256.0f / FOV_H;
constexpr int   BLK = 256;

typedef __attribute__((ext_vector_type(2))) float f2;
typedef __attribute__((ext_vector_type(4))) float f4;

struct V3 { float x, y, z; };

__device__ __forceinline__ V3 cross3(V3 a, V3 b) {
  return { a.y * b.z - a.z * b.y,
           a.z * b.x - a.x * b.z,
           a.x * b.y - a.y * b.x };
}
__device__ __forceinline__ V3 add3(V3 a, V3 b) { return { a.x + b.x, a.y + b.y, a.z + b.z }; }
__device__ __forceinline__ V3 sub3(V3 a, V3 b) { return { a.x - b.x, a.y - b.y, a.z - b.z }; }
__device__ __forceinline__ V3 scal3(float s, V3 a) { return { s * a.x, s * a.y, s * a.z }; }
__device__ __forceinline__ float dot3(V3 a, V3 b) { return a.x * b.x + a.y * b.y + a.z * b.z; }

// quat_rot: p + w*t + v×t, with t = 2*(v×p)
__device__ __forceinline__ V3 quat_rot(V3 v, float w, V3 p) {
  V3 t = scal3(2.0f, cross3(v, p));
  return add3(add3(p, scal3(w, t)), cross3(v, t));
}

} // namespace

// ---------------------------------------------------------------------------
// Main kernel: one thread per observation e.
//  - Pose table (2048 poses) staged into LDS via CDNA5 async global->LDS
//    copies (ASYNCcnt path) into a PADDED stride-32B layout, so each random
//    pose gather is exactly two aligned ds_load_b128.
//  - Streaming (touch-once) traffic uses non-temporal hints; the L2-resident
//    patch/elevation gather tables keep regular temporal hints.
// ---------------------------------------------------------------------------
__global__ __launch_bounds__(BLK) void ba_proj_elev_kernel(
    const float* __restrict__ poses,
    const float* __restrict__ elev,
    const float* __restrict__ patch_coords,
    const float* __restrict__ target_coords,
    const float* __restrict__ init_elev,
    const int*   __restrict__ source_idx,
    const int*   __restrict__ target_idx,
    const int*   __restrict__ patch_idx,
    float* __restrict__ out_proj,   // 2*E_NUM, interleaved (err_r, err_t)
    float* __restrict__ out_elev)   // E_NUM
{
  // stride 8 floats per pose (floats 0..6 valid, 7 = pad) -> 64 KB
  __shared__ alignas(16) float s_poses[P_NUM * 8];

  // ---- async stage: 3 async ops per pose; inst offset applies to both the
  //      global and LDS address, and the pad sits at the tail, so offsets match.
  {
    const unsigned lds_base = (unsigned)(unsigned long long)(&s_poses[0]);
#pragma unroll
    for (int k = 0; k < P_NUM / BLK; ++k) {
      const int p = (int)threadIdx.x + k * BLK;
      const float* g = poses + p * POSE_F;
      const unsigned l = lds_base + (unsigned)p * 32u;
      asm volatile("global_load_async_to_lds_b128 %0, %1, off"
                   :: "v"(l), "v"(g) : "memory");
      asm volatile("global_load_async_to_lds_b64 %0, %1, off offset:16"
                   :: "v"(l), "v"(g) : "memory");
      asm volatile("global_load_async_to_lds_b32 %0, %1, off offset:24"
                   :: "v"(l), "v"(g) : "memory");
    }
    asm volatile("s_wait_asynccnt 0" ::: "memory");
  }
  __syncthreads();

  const int e = (int)blockIdx.x * BLK + (int)threadIdx.x;

  // streaming index gathers (touch-once -> non-temporal)
  const int si = __builtin_nontemporal_load(source_idx + e);
  const int ti = __builtin_nontemporal_load(target_idx + e);
  const int pi = __builtin_nontemporal_load(patch_idx + e);

  // polar2cart of [r, th, ph] (patch tables are L2-resident with reuse -> RT)
  const f2 pc = ((const f2*)patch_coords)[pi];
  const float r  = pc.x;
  const float th = pc.y;
  const float ph = elev[pi];

  float sth, cth, sph, cph;
  sincosf(th, &sth, &cth);
  sincosf(ph, &sph, &cph);
  const V3 p = { r * cph * cth, r * cph * sth, r * sph };

  // global_pt = se3_act(sp, p)   (two aligned ds_load_b128)
  const f4* spv = reinterpret_cast<const f4*>(&s_poses[si * 8]);
  const f4 sA = spv[0];               // t0 t1 t2 qx
  const f4 sB = spv[1];               // qy qz qw pad
  const V3 st = { sA.x, sA.y, sA.z };
  const V3 sv = { sA.w, sB.x, sB.y };
  const float sw = sB.z;
  const V3 g = add3(quat_rot(sv, sw, p), st);

  // local_tgt = se3_act(se3_inv(tp), global_pt)
  const f4* tpv = reinterpret_cast<const f4*>(&s_poses[ti * 8]);
  const f4 tA = tpv[0];
  const f4 tB = tpv[1];
  const V3 tt = { tA.x, tA.y, tA.z };
  const V3 iv = { -tA.w, -tB.x, -tB.y };
  const float iw = tB.z;
  const V3 tinv = scal3(-1.0f, quat_rot(iv, iw, tt));
  const V3 lt   = add3(quat_rot(iv, iw, g), tinv);

  // cart2polar (only r and theta are consumed)
  const float rr  = sqrtf(dot3(lt, lt));
  const float thp = atan2f(lt.y, lt.x);

  const f2 tc = __builtin_nontemporal_load(((const f2*)target_coords) + e);
  f2 res;
  res.x = (rr  - tc.x) * R_SCALE;
  res.y = (thp - tc.y) * T_SCALE;
  __builtin_nontemporal_store(res, ((f2*)out_proj) + e);

  const float ev = elev[e] - __builtin_nontemporal_load(init_elev + e);
  __builtin_nontemporal_store(ev, out_elev + e);
}

// ---------------------------------------------------------------------------
// Pose residual: one thread per pose -> se3_log(se3_mul(se3_inv(init), pose))
// ---------------------------------------------------------------------------
__global__ __launch_bounds__(BLK) void ba_pose_kernel(
    const float* __restrict__ poses,
    const float* __restrict__ init_poses,
    float* __restrict__ out_pose)   // 6 * P_NUM
{
  const int pid = (int)blockIdx.x * BLK + (int)threadIdx.x;
  if (pid >= P_NUM) return;

  const float* a = init_poses + pid * POSE_F;
  const float* b = poses      + pid * POSE_F;

  const V3 at = { a[0], a[1], a[2] };
  const V3 av = { a[3], a[4], a[5] };  const float aw = a[6];
  const V3 bt = { b[0], b[1], b[2] };
  const V3 bv = { b[3], b[4], b[5] };  const float bw = b[6];

  // se3_inv(a)
  const V3 qiv = scal3(-1.0f, av);
  const float qiw = aw;
  const V3 tinv = scal3(-1.0f, quat_rot(qiv, qiw, at));

  // rel = se3_mul(inv(a), b)
  const V3 t = add3(quat_rot(qiv, qiw, bt), tinv);
  float w = qiw * bw - dot3(qiv, bv);
  V3 v = add3(add3(scal3(qiw, bv), scal3(bw, qiv)), cross3(qiv, bv));

  // so3_log
  const float s = (w < 0.0f) ? -1.0f : 1.0f;
  v = scal3(s, v);
  w = s * w;
  const float n = sqrtf(dot3(v, v));
  const float theta = 2.0f * atan2f(n, w);
  const float scale = (n < 1e-7f) ? (2.0f / fmaxf(w, 1e-7f))
                                  : (theta / fmaxf(n, 1e-12f));
  const V3 phi = scal3(scale, v);

  // se3_log
  const float thn = sqrtf(dot3(phi, phi));
  const float th2 = thn * thn;
  const float coef = (thn < 1e-4f)
      ? (1.0f / 12.0f)
      : (1.0f / fmaxf(th2, 1e-12f)
         - (1.0f + cosf(thn)) / (2.0f * fmaxf(thn * sinf(thn), 1e-12f)));
  const V3 cp  = cross3(phi, t);
  const V3 rho = add3(sub3(t, scal3(0.5f, cp)), scal3(coef, cross3(phi, cp)));

  float* o = out_pose + pid * 6;
  o[0] = rho.x; o[1] = rho.y; o[2] = rho.z;
  o[3] = phi.x; o[4] = phi.y; o[5] = phi.z;
}

// ---------------------------------------------------------------------------
extern "C" void kernel_launch(void* const* d_in, const int* in_sizes, int n_in,
                              void* d_out, int out_size, void* d_ws, size_t ws_size,
                              hipStream_t stream) {
  (void)in_sizes; (void)n_in; (void)out_size; (void)d_ws; (void)ws_size;

  const float* poses        = (const float*)d_in[0];
  const float* elev         = (const float*)d_in[1];
  const float* patch_coords = (const float*)d_in[2];
  const float* target_coords= (const float*)d_in[3];
  const float* init_poses   = (const float*)d_in[4];
  const float* init_elev    = (const float*)d_in[5];
  const int*   source_idx   = (const int*)d_in[6];
  const int*   target_idx   = (const int*)d_in[7];
  const int*   patch_idx    = (const int*)d_in[8];

  float* out       = (float*)d_out;
  float* out_proj  = out;                                  // 2*E_NUM
  float* out_pose  = out + (size_t)2 * E_NUM;              // 6*P_NUM
  float* out_elev  = out_pose + (size_t)6 * P_NUM;         // E_NUM

  ba_proj_elev_kernel<<<E_NUM / BLK, BLK, 0, stream>>>(
      poses, elev, patch_coords, target_coords, init_elev,
      source_idx, target_idx, patch_idx, out_proj, out_elev);

  ba_pose_kernel<<<(P_NUM + BLK - 1) / BLK, BLK, 0, stream>>>(
      poses, init_poses, out_pose);
}